// ODEFunction_76295799046809
// MI455X (gfx1250) — compile-verified
//
#include <hip/hip_runtime.h>
#include <hip/hip_bf16.h>

// SpMM (COO, rows sorted) via WMMA-based segmented reduction.
//   out[r, :] = sum_e (rows[e]==r) vals[e] * x[cols[e], :]
//
// rows sorted => edges for the 16-row window [16w, 16w+16) are contiguous.
// Kernel 1 binary-searches window start offsets. Kernel 2: one wave32 per
// window; segment sum computed as C += A*B with V_WMMA_F32_16X16X4_F32,
// A = 16x4 one-hot row-selection matrix scaled by vals, B = 4x16 gathered
// feature tile (x is 19.2 MB -> L2 resident). 48 feats = 3 accumulator
// tiles. No atomics; each window's 16x48 output owned by one wave. Full f32.
//
// v3: uniform-guard store epilogue (single base address + immediate-offset
// store clause) replacing 16 exec-masked per-row guards.

typedef __attribute__((ext_vector_type(2))) float v2f;
typedef __attribute__((ext_vector_type(8))) float v8f;

#define D_FEAT 48
#define ROWS_PER_WIN 16

__global__ void win_start_kernel(const int* __restrict__ rows,
                                 int* __restrict__ ws,
                                 int n_edges, int n_win) {
    int w = blockIdx.x * blockDim.x + threadIdx.x;
    if (w > n_win) return;                 // n_win+1 entries (sentinel at end)
    int target = w * ROWS_PER_WIN;
    int lo = 0, hi = n_edges;
    while (lo < hi) {                      // lower_bound(rows, target)
        int mid = (lo + hi) >> 1;
        if (rows[mid] < target) lo = mid + 1; else hi = mid;
    }
    ws[w] = lo;
}

__global__ __launch_bounds__(32) void spmm_wmma_kernel(
    const float* __restrict__ x,
    const int* __restrict__ rows,
    const int* __restrict__ cols,
    const float* __restrict__ vals,
    const int* __restrict__ ws,
    float* __restrict__ out,
    int n_nodes) {
    const int w    = blockIdx.x;
    const int lane = threadIdx.x;      // wave32
    const int m    = lane & 15;        // row (A) / col (B,C) within 16-tile
    const int half = lane >> 4;        // K-half selector per ISA VGPR layout
    const int rbase   = w * ROWS_PER_WIN;
    const int e_begin = ws[w];
    const int e_end   = ws[w + 1];
    const int last    = e_end - 1;     // valid whenever the loop is entered
    const int rm      = rbase + m;     // row this lane selects for in A

    v8f acc0 = {}; v8f acc1 = {}; v8f acc2 = {};

#pragma unroll 2
    for (int e = e_begin; e < e_end; e += 4) {
        // ISA 16x4 f32 A layout: lanes 0-15 hold K=0 (v0), K=1 (v1);
        // lanes 16-31 hold K=2 (v0), K=3 (v1). B (4x16) mirrors per-VGPR.
        const int ea  = e + 2 * half;
        const int eb  = ea + 1;
        // Branch-free tail handling: clamp index (always in-range), mask value.
        const int eac = ea < last ? ea : last;
        const int ebc = eb < last ? eb : last;

        const float va = vals[eac];
        const float vb = vals[ebc];
        const int   ra = rows[eac];
        const int   rb = rows[ebc];
        const int   oa = cols[eac] * D_FEAT;   // 32-bit element offsets
        const int   ob = cols[ebc] * D_FEAT;

        // A: one-hot (row match within window) scaled by edge weight;
        // out-of-range group slots contribute exactly zero.
        v2f A;
        A.x = (ea <= last && ra == rm) ? va : 0.0f;
        A.y = (eb <= last && rb == rm) ? vb : 0.0f;

        // B: gathered feature tiles x[col, nb + m] (L2 hits)
        v2f B0, B1, B2;
        B0.x = x[oa + m];      B0.y = x[ob + m];
        B1.x = x[oa + 16 + m]; B1.y = x[ob + 16 + m];
        B2.x = x[oa + 32 + m]; B2.y = x[ob + 32 + m];

        // D = A x B + C  (3 independent accumulator chains for ILP)
        acc0 = __builtin_amdgcn_wmma_f32_16x16x4_f32(
            false, A, false, B0, (short)0, acc0, false, false);
        acc1 = __builtin_amdgcn_wmma_f32_16x16x4_f32(
            false, A, false, B1, (short)0, acc1, false, false);
        acc2 = __builtin_amdgcn_wmma_f32_16x16x4_f32(
            false, A, false, B2, (short)0, acc2, false, false);
    }

    // C/D layout: VGPR j holds row M = j (lanes 0-15) / M = j+8 (lanes 16-31),
    // N = m. One base address; 24 stores with immediate offsets (clauseable),
    // coalesced 64B per 16 lanes.
    const int row0 = rbase + 8 * half;
    float* o = out + (size_t)row0 * D_FEAT + m;
    if (rbase + ROWS_PER_WIN <= n_nodes) {   // uniform (scalar) branch
#pragma unroll
        for (int j = 0; j < 8; ++j) {
            o[j * D_FEAT +  0] = acc0[j];
            o[j * D_FEAT + 16] = acc1[j];
            o[j * D_FEAT + 32] = acc2[j];
        }
    } else {                                  // generality: ragged last window
#pragma unroll
        for (int j = 0; j < 8; ++j) {
            if (row0 + j < n_nodes) {
                o[j * D_FEAT +  0] = acc0[j];
                o[j * D_FEAT + 16] = acc1[j];
                o[j * D_FEAT + 32] = acc2[j];
            }
        }
    }
}

extern "C" void kernel_launch(void* const* d_in, const int* in_sizes, int n_in,
                              void* d_out, int out_size, void* d_ws, size_t ws_size,
                              hipStream_t stream) {
    // inputs: t(f32,1), x(f32,N*48), rows(i32,E), cols(i32,E), vals(f32,E)
    const float* x    = (const float*)d_in[1];
    const int*   rows = (const int*)d_in[2];
    const int*   cols = (const int*)d_in[3];
    const float* vals = (const float*)d_in[4];
    float*       out  = (float*)d_out;

    const int n_edges = in_sizes[2];
    const int n_nodes = in_sizes[1] / D_FEAT;
    const int n_win   = (n_nodes + ROWS_PER_WIN - 1) / ROWS_PER_WIN;

    int* ws = (int*)d_ws;  // n_win+1 window start offsets

    win_start_kernel<<<(n_win + 1 + 255) / 256, 256, 0, stream>>>(
        rows, ws, n_edges, n_win);
    spmm_wmma_kernel<<<n_win, 32, 0, stream>>>(
        x, rows, cols, vals, ws, out, n_nodes);
}